// VectorQuantizer_54924041781316
// MI455X (gfx1250) — compile-verified
//
#include <hip/hip_runtime.h>
#include <hip/hip_bf16.h>
#include <math.h>

typedef __attribute__((ext_vector_type(2))) float v2f;
typedef __attribute__((ext_vector_type(8))) float v8f;
typedef __attribute__((ext_vector_type(4))) unsigned int v4u;
typedef __attribute__((ext_vector_type(8))) int v8i;
typedef __attribute__((ext_vector_type(4))) int v4i;

#define N_INPUTS 32768
#define NUM_EMB  4096
#define DIM      256

// ws layout (bytes):
//   [0,      16384)   e_norm      float[4096]
//   [16384, 147456)   idx         int[32768]
//   [147456,163840)   count       int[4096]
//   [163840,196608)   lossPartial float[8192]

// ---------------------------------------------------------------------------
// Kernel 1: per-codebook-row squared norm, one wave32 per row
// ---------------------------------------------------------------------------
__global__ void __launch_bounds__(256) vq_enorm(const float* __restrict__ weight,
                                                float* __restrict__ enorm) {
    int wid  = (blockIdx.x * blockDim.x + threadIdx.x) >> 5;   // row id
    int lane = threadIdx.x & 31;
    const float* wr = weight + (size_t)wid * DIM;
    float s = 0.f;
#pragma unroll
    for (int i = 0; i < 8; ++i) {
        float w = wr[lane + i * 32];
        s += w * w;
    }
#pragma unroll
    for (int off = 16; off >= 1; off >>= 1)
        s += __shfl_xor(s, off, 32);
    if (lane == 0) enorm[wid] = s;
}

// ---------------------------------------------------------------------------
// Kernel 2: WMMA distance GEMM + running argmin.
// Block = 256 threads (8 waves); wave 0 DMAs the block's 128x256 f32 input
// tile into LDS with the Tensor Data Mover (TENSORcnt), then each wave
// computes a 16(M) x 4096(N) score strip with K=256 using
// v_wmma_f32_16x16x4_f32 with 4 accumulators per A fragment.
// ---------------------------------------------------------------------------
__global__ void __launch_bounds__(256) vq_argmin(const float* __restrict__ inputs,
                                                 const float* __restrict__ weight,
                                                 const float* __restrict__ enorm,
                                                 int* __restrict__ idx,
                                                 int* __restrict__ count) {
    extern __shared__ float lds[];   // 128 rows x 256 f32 = 128 KB

    // --- TDM stage of the A tile: one tensor_load_to_lds from wave 0 -------
    if (threadIdx.x < 32) {
        unsigned long long ga =
            (unsigned long long)(inputs + (size_t)blockIdx.x * 128 * DIM);
        unsigned int ldsoff = (unsigned int)(size_t)lds;   // LDS byte offset

        v4u g0;
        g0.x = 1u;                                   // count=1, user descriptor
        g0.y = ldsoff;                               // lds_addr [63:32]
        g0.z = (unsigned int)(ga & 0xFFFFFFFFu);     // global_addr [95:64]
        g0.w = (unsigned int)((ga >> 32) & 0x01FFFFFFu) | (2u << 30); // addr hi | type=2

        v8i g1;
        g1[0] = 0x00020000;   // workgroup_mask=0, data_size=2 (4 bytes)
        g1[1] = 0x01000000;   // tensor_dim0 = 256  (bits [79:48], low half)
        g1[2] = 0x00800000;   // tensor_dim1 = 128  (bits [111:80])
        g1[3] = 0x01000000;   // tile_dim0   = 256  (bits [127:112])
        g1[4] = 0x00000080;   // tile_dim1   = 128  (bits [143:128]); tile_dim2=0
        g1[5] = 0x00000100;   // tensor_dim0_stride = 256 (bits [207:160])
        g1[6] = 0;
        g1[7] = 0;

        v4i gz  = {0, 0, 0, 0};            // groups 2/3 unused for 2-D tile
        v8i gz8 = {0, 0, 0, 0, 0, 0, 0, 0};

        __builtin_amdgcn_tensor_load_to_lds(g0, g1, gz, gz, gz8, 0);
        __builtin_amdgcn_s_wait_tensorcnt(0);
    }
    __syncthreads();

    const int wave = threadIdx.x >> 5;
    const int lane = threadIdx.x & 31;
    const int half = lane >> 4;      // selects K-pair (A) / K-pair (B)
    const int nrow = lane & 15;      // M row (A) / N col (B)

    // A fragment base: lane holds A[M=nrow, K = s*4 + half*2 + {0,1}]
    const float* abase = lds + (size_t)((wave << 4) + nrow) * DIM + half * 2;

    float minval[8];
    int   minidx[8];
#pragma unroll
    for (int v = 0; v < 8; ++v) { minval[v] = 3.4e38f; minidx[v] = 0; }

    for (int nb = 0; nb < 64; ++nb) {          // 64-column blocks of N
        const int n0 = nb * 64;
        // B fragment: lane holds B[K = s*4 + half*2 + {0,1}, N = n0 + j*16 + nrow]
        //           = weight[(n0 + j*16 + nrow)][k..k+1]
        const float* wb = weight + (size_t)(n0 + nrow) * DIM + half * 2;

        v8f acc0 = {}, acc1 = {}, acc2 = {}, acc3 = {};
#pragma unroll 4
        for (int s = 0; s < 64; ++s) {
            v2f a  = *(const v2f*)(abase + s * 4);
            v2f b0 = *(const v2f*)(wb + s * 4);
            v2f b1 = *(const v2f*)(wb + 16 * DIM + s * 4);
            v2f b2 = *(const v2f*)(wb + 32 * DIM + s * 4);
            v2f b3 = *(const v2f*)(wb + 48 * DIM + s * 4);
            acc0 = __builtin_amdgcn_wmma_f32_16x16x4_f32(false, a, false, b0, (short)0, acc0, false, false);
            acc1 = __builtin_amdgcn_wmma_f32_16x16x4_f32(false, a, false, b1, (short)0, acc1, false, false);
            acc2 = __builtin_amdgcn_wmma_f32_16x16x4_f32(false, a, false, b2, (short)0, acc2, false, false);
            acc3 = __builtin_amdgcn_wmma_f32_16x16x4_f32(false, a, false, b3, (short)0, acc3, false, false);
        }

        // score = |e_n|^2 - 2 * dot ; fold into running per-row argmin.
        // C/D layout: VGPR v holds (M = v + 8*half, N = n0 + j*16 + nrow).
#pragma unroll
        for (int j = 0; j < 4; ++j) {
            v8f acc = (j == 0) ? acc0 : (j == 1) ? acc1 : (j == 2) ? acc2 : acc3;
            int n = n0 + j * 16 + nrow;
            float en = enorm[n];
#pragma unroll
            for (int v = 0; v < 8; ++v) {
                float score = en - 2.0f * acc[v];
                if (score < minval[v]) { minval[v] = score; minidx[v] = n; }
            }
        }
    }

    // reduce (min, argmin) across the 16 lanes of each half-wave,
    // tie-break toward the lower index (matches jnp.argmin).
#pragma unroll
    for (int off = 8; off >= 1; off >>= 1) {
#pragma unroll
        for (int v = 0; v < 8; ++v) {
            float ov = __shfl_xor(minval[v], off, 16);
            int   oi = __shfl_xor(minidx[v], off, 16);
            if (ov < minval[v] || (ov == minval[v] && oi < minidx[v])) {
                minval[v] = ov; minidx[v] = oi;
            }
        }
    }

    if (nrow == 0) {   // lane 0 -> rows m=0..7, lane 16 -> rows m=8..15
        int row0 = blockIdx.x * 128 + (wave << 4) + (half << 3);
#pragma unroll
        for (int v = 0; v < 8; ++v) {
            idx[row0 + v] = minidx[v];
            atomicAdd(&count[minidx[v]], 1);
        }
    }
}

// ---------------------------------------------------------------------------
// Kernel 3: gather quantized = weight[idx] (== quantized_st numerically)
// plus per-block sum of squared error for the loss. 4 rows / 256-thread block.
// qout is only 4-byte aligned (d_out + 1) -> scalar stores.
// ---------------------------------------------------------------------------
__global__ void __launch_bounds__(256) vq_gather(const float* __restrict__ inputs,
                                                 const float* __restrict__ weight,
                                                 const int* __restrict__ idx,
                                                 float* __restrict__ qout,
                                                 float* __restrict__ lossPartial) {
    __shared__ float red[256];
    int t   = threadIdx.x;
    int row = blockIdx.x * 4 + (t >> 6);
    int c   = t & 63;
    int k   = idx[row];
    float4 w4 = ((const float4*)weight)[(size_t)k * 64 + c];
    float4 x4 = ((const float4*)inputs)[(size_t)row * 64 + c];
    size_t o = (size_t)row * DIM + (size_t)c * 4;
    qout[o + 0] = w4.x; qout[o + 1] = w4.y; qout[o + 2] = w4.z; qout[o + 3] = w4.w;
    float dx = w4.x - x4.x, dy = w4.y - x4.y, dz = w4.z - x4.z, dw = w4.w - x4.w;
    red[t] = dx * dx + dy * dy + dz * dz + dw * dw;
    __syncthreads();
    for (int off = 128; off >= 1; off >>= 1) {
        if (t < off) red[t] += red[t + off];
        __syncthreads();
    }
    if (t == 0) lossPartial[blockIdx.x] = red[0];
}

// ---------------------------------------------------------------------------
// Kernel 4: one-hot encodings, one block per row; float2 (b64) stores
// (enc base = d_out + 8388610 floats -> 8-byte aligned, not 16).
// ---------------------------------------------------------------------------
__global__ void __launch_bounds__(1024) vq_encodings(const int* __restrict__ idx,
                                                     float2* __restrict__ enc) {
    int r = blockIdx.x;
    int k = idx[r];
    int t = threadIdx.x;
    size_t base = (size_t)r * 2048;   // 4096 floats = 2048 float2
#pragma unroll
    for (int i = 0; i < 2; ++i) {
        int c = t + i * 1024;
        float2 val;
        val.x = ((k >> 1) == c && (k & 1) == 0) ? 1.0f : 0.0f;
        val.y = ((k >> 1) == c && (k & 1) == 1) ? 1.0f : 0.0f;
        enc[base + c] = val;
    }
}

// ---------------------------------------------------------------------------
// Kernel 5: finalize loss (1.25 * mean sq err) and perplexity from histogram.
// ---------------------------------------------------------------------------
__global__ void __launch_bounds__(1024) vq_finalize(const float* __restrict__ lossPartial,
                                                    const int* __restrict__ count,
                                                    float* __restrict__ out_loss,
                                                    float* __restrict__ out_perp) {
    __shared__ float red[1024];
    int t = threadIdx.x;

    float s = 0.f;
    for (int i = t; i < 8192; i += 1024) s += lossPartial[i];
    red[t] = s;
    __syncthreads();
    for (int off = 512; off >= 1; off >>= 1) {
        if (t < off) red[t] += red[t + off];
        __syncthreads();
    }
    if (t == 0)
        out_loss[0] = red[0] * 1.25f / ((float)N_INPUTS * (float)DIM);
    __syncthreads();

    float p = 0.f;
    for (int i = t; i < NUM_EMB; i += 1024) {
        float pr = (float)count[i] * (1.0f / (float)N_INPUTS);
        p += pr * __logf(pr + 1e-10f);
    }
    red[t] = p;
    __syncthreads();
    for (int off = 512; off >= 1; off >>= 1) {
        if (t < off) red[t] += red[t + off];
        __syncthreads();
    }
    if (t == 0)
        out_perp[0] = __expf(-red[0]);
}

// ---------------------------------------------------------------------------
extern "C" void kernel_launch(void* const* d_in, const int* in_sizes, int n_in,
                              void* d_out, int out_size, void* d_ws, size_t ws_size,
                              hipStream_t stream) {
    const float* inputs = (const float*)d_in[0];   // [32768, 256]
    const float* weight = (const float*)d_in[1];   // [4096, 256]
    float* out = (float*)d_out;
    char*  ws  = (char*)d_ws;

    float* enorm       = (float*)(ws);
    int*   idx         = (int*)(ws + 16384);
    int*   count       = (int*)(ws + 147456);
    float* lossPartial = (float*)(ws + 163840);

    // output layout: [loss(1) | quantized_st(32768*256) | perplexity(1) | encodings(32768*4096)]
    float*  q_out    = out + 1;
    float*  perp_out = out + 1 + (size_t)N_INPUTS * DIM;            // 8388609
    float2* enc_out  = (float2*)(out + 2 + (size_t)N_INPUTS * DIM); // 8388610 (8B aligned)

    (void)hipMemsetAsync(count, 0, NUM_EMB * sizeof(int), stream);

    vq_enorm    <<<NUM_EMB / 8, 256, 0, stream>>>(weight, enorm);
    vq_argmin   <<<N_INPUTS / 128, 256, 131072, stream>>>(inputs, weight, enorm, idx, count);
    vq_gather   <<<N_INPUTS / 4, 256, 0, stream>>>(inputs, weight, idx, q_out, lossPartial);
    vq_encodings<<<N_INPUTS, 1024, 0, stream>>>(idx, enc_out);
    vq_finalize <<<1, 1024, 0, stream>>>(lossPartial, count, out, perp_out);
}